// STLSTMLoRAInline_12438225289392
// MI455X (gfx1250) — compile-verified
//
#include <hip/hip_runtime.h>
#include <math.h>

typedef __attribute__((ext_vector_type(2))) float v2f;
typedef __attribute__((ext_vector_type(8))) float v8f;

#define RANK   8
#define SCALING 2.0f            // ALPHA/RANK = 16/8
#define BB     8
#define TT     16
#define PP     197
#define CC     1024
#define MTOT   (BB*TT*PP)       // 25216 rows, = 1576 tiles of 16
#define NSEQ   (BB*PP)          // 1576 LSTM sequences
#define KSPLIT 4                // K-chunks for the down GEMM
#define KCHUNK (CC/KSPLIT)      // 256
#define STAGE  32               // K columns staged into LDS per step
#define NSTAGE (KCHUNK/STAGE)   // 8

// ---- async global->LDS (ASYNCcnt path) via inline asm -----------------------
// VGLOBAL async encoding: vdst = VGPR holding wave-relative LDS byte offset,
// vaddr = 64-bit global address VGPR pair, "off" = no SGPR base.
__device__ __forceinline__ void async_b128(const float* g, float* l) {
  unsigned ldsaddr = (unsigned)(unsigned long long)(const void*)l;
  unsigned long long gaddr = (unsigned long long)(const void*)g;
  asm volatile("global_load_async_to_lds_b128 %0, %1, off"
               :: "v"(ldsaddr), "v"(gaddr) : "memory");
}

// Wait for ASYNCcnt <= imm and return a zero token as an opaque SGPR value.
// The LDS fragment addresses are offset by the token, creating a true data
// dependence: the scheduler cannot hoist the ds_loads above the wait (a plain
// "memory"-clobber barrier was observed to be crossed by the MI scheduler).
__device__ __forceinline__ unsigned wait_async_le4_tok() {
  unsigned t;
  asm volatile("s_wait_asynccnt 0x4\n\ts_mov_b32 %0, 0" : "=s"(t) :: "memory");
  return t;
}
__device__ __forceinline__ unsigned wait_async_le0_tok() {
  unsigned t;
  asm volatile("s_wait_asynccnt 0x0\n\ts_mov_b32 %0, 0" : "=s"(t) :: "memory");
  return t;
}
// WAR guard: all of this wave's outstanding LDS reads complete before the next
// async issue group may overwrite their source buffer.
#define WAIT_DS_DONE() asm volatile("s_wait_dscnt 0x0" ::: "memory")

// ---------------------------------------------------------------------------
// Kernel 1: LoRA down projection.  part[kc][m][r] = sum_{k in chunk} hs[m,k]*dw[r,k]
// One wave = one 16-row M tile x one 256-wide K chunk.  A panels (16x32 f32,
// 2 KB) are double-buffered in LDS via GLOBAL_LOAD_ASYNC_TO_LDS_B128; WMMA A
// fragments come from LDS (lane L, vgpr v holds A[L%16, k + 2*(L/16) + v]).
// B fragments come straight from dw (32 KB, cache resident); columns n>=8 read
// duplicate row 0 (finite garbage, never stored).
// ---------------------------------------------------------------------------
__global__ __launch_bounds__(256) void lora_down_kernel(
    const float* __restrict__ hs, const float* __restrict__ dw,
    float* __restrict__ part) {
  __shared__ float lds[8][2][16 * STAGE];   // 8 waves x 2 buffers x 2KB = 32KB

  const int lane  = threadIdx.x & 31;
  const int wib   = threadIdx.x >> 5;       // wave in block
  const int wave  = blockIdx.x * (blockDim.x >> 5) + wib;
  const int tile  = wave >> 2;              // M tile
  const int kc    = wave & (KSPLIT-1);      // K chunk
  if (tile >= MTOT/16) return;              // wave-uniform
  const int half  = lane >> 4;
  const int lrow  = lane & 15;
  const int mbase = tile * 16;
  const int kstart = kc * KCHUNK;

  // staging geometry: lane covers row (lane&15), segments (lane>>4)+2j, j=0..3
  const int srow = lane & 15;
  const int sseg = lane >> 4;
  const float* gstage = hs + (size_t)(mbase + srow) * CC + kstart + sseg * 4;
  float* l0 = &lds[wib][0][srow * STAGE + sseg * 4];
  float* l1 = &lds[wib][1][srow * STAGE + sseg * 4];
  // per-wave buffer bases for compute-side reads
  const float* rb0 = &lds[wib][0][lrow * STAGE + 2 * half];
  const float* rb1 = &lds[wib][1][lrow * STAGE + 2 * half];

  const int brow = (lrow < RANK) ? lrow : 0;   // clamp: in-bounds, harmless dup
  const float* bptr = dw + (size_t)brow * CC + kstart + 2*half;

  // prefetch stage 0 into buffer 0
  #pragma unroll
  for (int j = 0; j < 4; ++j) async_b128(gstage + j*8, l0 + j*8);

  v8f acc = {};
  #pragma unroll
  for (int s = 0; s < NSTAGE; ++s) {
    unsigned tok;
    if (s + 1 < NSTAGE) {                    // prefetch next stage
      WAIT_DS_DONE();                        // WAR: prior reads of target buffer done
      const float* gn = gstage + (size_t)(s + 1) * STAGE;
      float* ln = ((s + 1) & 1) ? l1 : l0;
      #pragma unroll
      for (int j = 0; j < 4; ++j) async_b128(gn + j*8, ln + j*8);
      tok = wait_async_le4_tok();            // current stage's 4 loads done
    } else {
      tok = wait_async_le0_tok();
    }
    const float* lb = ((s & 1) ? rb1 : rb0) + tok;   // token-carried dependence
    #pragma unroll
    for (int kk = 0; kk < STAGE; kk += 4) {
      v2f a = *(const v2f*)(lb + kk);        // ds_load_b64, WMMA A fragment
      v2f b = *(const v2f*)(bptr);
      acc = __builtin_amdgcn_wmma_f32_16x16x4_f32(false, a, false, b,
                                                  (short)0, acc, false, false);
      bptr += 4;
    }
  }

  // D layout: vgpr g, lane L -> D[M = g + 8*(L/16), N = L%16]
  if (lrow < RANK) {
    float* dst = part + (size_t)kc * (MTOT * RANK);
    #pragma unroll
    for (int g = 0; g < 8; ++g) {
      int m = mbase + g + 8*half;
      dst[(size_t)m * RANK + lrow] = acc[g];
    }
  }
}

// ---------------------------------------------------------------------------
// Kernel 2: temporal LSTM in rank space.  One wave per (b,p) sequence.
// Lane j (0..31) owns gate j (PyTorch order: rows 0-7=i, 8-15=f, 16-23=g, 24-31=o).
// h/c replicated across lanes via __shfl.  Sums the 4 down-partials and applies
// frame_mask on input; folds mask*SCALING into the stored h.
// ---------------------------------------------------------------------------
__device__ __forceinline__ float sigf(float x) { return 1.0f / (1.0f + __expf(-x)); }

__global__ __launch_bounds__(256) void lstm_kernel(
    const float* __restrict__ part,
    const float* __restrict__ wih, const float* __restrict__ whh,
    const float* __restrict__ bih, const float* __restrict__ bhh,
    const float* __restrict__ mask, float* __restrict__ hsout) {
  const int lane = threadIdx.x & 31;
  const int seq  = blockIdx.x * (blockDim.x >> 5) + (threadIdx.x >> 5);
  if (seq >= NSEQ) return;               // wave-uniform
  const int b = seq / PP, p = seq % PP;

  float wi[RANK], wh[RANK];
  #pragma unroll
  for (int r = 0; r < RANK; ++r) {
    wi[r] = wih[lane*RANK + r];
    wh[r] = whh[lane*RANK + r];
  }
  const float bias = bih[lane] + bhh[lane];

  float h[RANK], c[RANK];
  #pragma unroll
  for (int r = 0; r < RANK; ++r) { h[r] = 0.f; c[r] = 0.f; }

  for (int t = 0; t < TT; ++t) {
    const size_t idx = ((size_t)(b*TT + t) * PP + p) * RANK;
    const float mk = mask[b*TT + t];
    float x[RANK];
    #pragma unroll
    for (int r = 0; r < RANK; ++r) {
      float s = part[idx + r]
              + part[(size_t)1*MTOT*RANK + idx + r]
              + part[(size_t)2*MTOT*RANK + idx + r]
              + part[(size_t)3*MTOT*RANK + idx + r];
      x[r] = s * mk;
    }
    float gv = bias;
    #pragma unroll
    for (int r = 0; r < RANK; ++r) gv += x[r]*wi[r] + h[r]*wh[r];
    #pragma unroll
    for (int r = 0; r < RANK; ++r) {
      float ig = __shfl(gv, r,      32);
      float fg = __shfl(gv, 8 + r,  32);
      float gg = __shfl(gv, 16 + r, 32);
      float og = __shfl(gv, 24 + r, 32);
      float cn = sigf(fg)*c[r] + sigf(ig)*tanhf(gg);
      c[r] = cn;
      h[r] = sigf(og)*tanhf(cn);
    }
    const float ms = mk * SCALING;
    if (lane == 0) {
      float4 lo = make_float4(h[0]*ms, h[1]*ms, h[2]*ms, h[3]*ms);
      float4 hi = make_float4(h[4]*ms, h[5]*ms, h[6]*ms, h[7]*ms);
      *(float4*)(hsout + idx)     = lo;
      *(float4*)(hsout + idx + 4) = hi;
    }
  }
}

// ---------------------------------------------------------------------------
// Kernel 3: LoRA up projection.  delta[m,c] = sum_r hsout[m,r] * uw[c,r]
// (mask*SCALING already folded into hsout).  Wave owns 16-row M tile and 16 of
// the 64 N tiles.  K=8 -> two chained f32 WMMAs.  Pure streaming-write pass.
// ---------------------------------------------------------------------------
__global__ __launch_bounds__(256) void lora_up_kernel(
    const float* __restrict__ hsout, const float* __restrict__ uw,
    float* __restrict__ delta) {
  const int lane = threadIdx.x & 31;
  const int wave = blockIdx.x * (blockDim.x >> 5) + (threadIdx.x >> 5);
  const int tile = wave >> 2;
  const int ng   = wave & 3;            // which quarter of the 64 N tiles
  if (tile >= MTOT/16) return;          // wave-uniform
  const int half = lane >> 4;
  const int lrow = lane & 15;
  const int mbase = tile * 16;

  const float* ap = hsout + (size_t)(mbase + lrow) * RANK + 2*half;
  v2f a0 = *(const v2f*)ap;         // k = 2*half + {0,1}
  v2f a1 = *(const v2f*)(ap + 4);   // k = 4 + 2*half + {0,1}

  for (int nt = ng*16; nt < ng*16 + 16; ++nt) {
    const int cbase = nt * 16;
    const float* bp = uw + (size_t)(cbase + lrow) * RANK + 2*half;
    v2f b0 = *(const v2f*)bp;       // B[k, n=lrow] = uw[c, r=k]
    v2f b1 = *(const v2f*)(bp + 4);
    v8f acc = {};
    acc = __builtin_amdgcn_wmma_f32_16x16x4_f32(false, a0, false, b0,
                                                (short)0, acc, false, false);
    acc = __builtin_amdgcn_wmma_f32_16x16x4_f32(false, a1, false, b1,
                                                (short)0, acc, false, false);
    #pragma unroll
    for (int g = 0; g < 8; ++g) {
      int m = mbase + g + 8*half;
      delta[(size_t)m * CC + cbase + lrow] = acc[g];
    }
  }
}

// ---------------------------------------------------------------------------
extern "C" void kernel_launch(void* const* d_in, const int* in_sizes, int n_in,
                              void* d_out, int out_size, void* d_ws, size_t ws_size,
                              hipStream_t stream) {
  const float* hs   = (const float*)d_in[0];  // (B,T,P,C)
  const float* mask = (const float*)d_in[2];  // (B,T)
  const float* dw   = (const float*)d_in[3];  // (R,C)
  const float* wih  = (const float*)d_in[4];  // (4R,R)
  const float* whh  = (const float*)d_in[5];  // (4R,R)
  const float* bih  = (const float*)d_in[6];  // (4R)
  const float* bhh  = (const float*)d_in[7];  // (4R)
  const float* uw   = (const float*)d_in[8];  // (C,R)
  float* out = (float*)d_out;                 // (B,T,P,C)

  float* part  = (float*)d_ws;                          // KSPLIT * MTOT * RANK
  float* hsv   = part + (size_t)KSPLIT * MTOT * RANK;   // MTOT * RANK

  const int waves_gemm = (MTOT/16) * KSPLIT;               // 6304 waves
  const int blocks_gemm = (waves_gemm * 32 + 255) / 256;   // 788
  const int blocks_lstm = (NSEQ * 32 + 255) / 256;         // 197

  lora_down_kernel<<<blocks_gemm, 256, 0, stream>>>(hs, dw, part);
  lstm_kernel<<<blocks_lstm, 256, 0, stream>>>(part, wih, whh, bih, bhh, mask, hsv);
  lora_up_kernel<<<blocks_gemm, 256, 0, stream>>>(hsv, uw, out);
}